// CrossAttention_46540265619919
// MI455X (gfx1250) — compile-verified
//
#include <hip/hip_runtime.h>

typedef __attribute__((ext_vector_type(2))) float v2f;
typedef __attribute__((ext_vector_type(8))) float v8f;

// D = A(16x4, f32) * B(4x16, f32) + C(16x16, f32)
static __device__ __forceinline__ v8f wmma4(v2f a, v2f b, v8f c) {
    return __builtin_amdgcn_wmma_f32_16x16x4_f32(
        /*neg_a=*/false, a, /*neg_b=*/false, b,
        /*c_mod=*/(short)0, c, /*reuse_a=*/false, /*reuse_b=*/false);
}

// ---------------------------------------------------------------------------
// Kernel 1: gene-axis reduction.  in: [rows, G, 64] f32 -> out: [rows, 64].
// One block per row; fully-coalesced float4 streaming (this is the HBM-bound
// part: 256 MiB + 128 MiB of input traffic).
// ---------------------------------------------------------------------------
__global__ __launch_bounds__(256) void reduce_genes(
    const float* __restrict__ in, float* __restrict__ out, int G) {
    const int blk = blockIdx.x;
    const int tid = threadIdx.x;
    const int d4 = tid & 15;   // which float4 of the 64-float feature row
    const int gs = tid >> 4;   // gene-slice 0..15

    const float4* p = (const float4*)(in + (size_t)blk * G * 64);
    float4 acc; acc.x = 0.f; acc.y = 0.f; acc.z = 0.f; acc.w = 0.f;
    for (int g = gs; g < G; g += 16) {
        float4 v = p[g * 16 + d4];
        acc.x += v.x; acc.y += v.y; acc.z += v.z; acc.w += v.w;
    }

    __shared__ float4 red[256];
    red[tid] = acc;
    __syncthreads();
    for (int s = 128; s >= 16; s >>= 1) {
        if (tid < s) {
            float4 o = red[tid + s];
            red[tid].x += o.x; red[tid].y += o.y;
            red[tid].z += o.z; red[tid].w += o.w;
        }
        __syncthreads();
    }
    if (tid < 16) ((float4*)(out + (size_t)blk * 64))[tid] = red[tid];
}

// ---------------------------------------------------------------------------
// Kernel 2: fused causal attention + residual + LayerNorm.
// Grid: B * (L/16) blocks, 256 threads (8 waves of 32).
// ---------------------------------------------------------------------------
#define LN_EPS 1.0e-3f
#define SM_STRIDE 516   // 516 % 64 == 4 -> conflict-free LDS rows

__global__ __launch_bounds__(256) void attn_fused(
    const float* __restrict__ x,      // [B, L, 512]
    const float* __restrict__ cv,     // [B, 512, 512] context_value
    const float* __restrict__ qred,   // [B, L, 64]
    const float* __restrict__ kred,   // [B, 512, 64]
    const float* __restrict__ gamma,  // [512]
    const float* __restrict__ beta,   // [512]
    float* __restrict__ out) {        // [B, L, 512]
    const int bid  = blockIdx.x;
    const int b    = bid >> 5;          // 32 L-tiles per batch
    const int lt   = bid & 31;
    const int l0   = lt << 4;
    const int tid  = threadIdx.x;
    const int w    = tid >> 5;          // wave 0..7
    const int lane = tid & 31;
    const int m    = lane & 15;         // row (A) / column N (B,C) index
    const int h    = lane >> 4;         // lane half

    __shared__ float sm[16 * SM_STRIDE];   // scores/P, later y-tile
    __shared__ float s_rowsum[16];

    // ---- preload A operand (q rows for this 16-row L tile) ----------------
    // ISA 32-bit A 16x4 layout: lane (h*16+m) holds K = {2h, 2h+1} per step.
    const float* qrow = qred + (size_t)(b * 512 + l0 + m) * 64;
    v2f aq[16];
#pragma unroll
    for (int s = 0; s < 16; ++s)
        aq[s] = *(const v2f*)(qrow + 4 * s + 2 * h);

    const int nk = lt + 1;                       // causal: K tiles 0..lt
    const float* kb = kred + (size_t)b * 512 * 64;

    // ---- Phase 1: scores = q_red . k_red^T (waves split K tiles) ----------
    for (int kt = w; kt < nk; kt += 8) {
        const int k0 = kt << 4;
        const float* krow = kb + (size_t)(k0 + m) * 64;  // B: N = lane&15
        v8f c = {};
#pragma unroll
        for (int s = 0; s < 16; ++s) {
            v2f bv = *(const v2f*)(krow + 4 * s + 2 * h);
            c = wmma4(aq[s], bv, c);
        }
        if (kt == lt) {                 // diagonal tile: mask k > l
#pragma unroll
            for (int r = 0; r < 8; ++r) {
                const int mr = r + 8 * h;       // C-layout row
                if (m > mr) c[r] -= 1.0e9f;     // N = m is the k column
            }
        }
#pragma unroll
        for (int r = 0; r < 8; ++r)
            sm[(r + 8 * h) * SM_STRIDE + k0 + m] = c[r];
    }
    __syncthreads();

    // ---- Phase 2: row softmax (16 lanes per row, half-wave shuffles) ------
    {
        const int row = tid >> 4;
        const int t16 = tid & 15;
        const int Kv  = nk << 4;
        float* prow = sm + row * SM_STRIDE;
        float mx = -3.0e38f;
        for (int k = t16; k < Kv; k += 16) mx = fmaxf(mx, prow[k]);
        for (int off = 8; off > 0; off >>= 1)
            mx = fmaxf(mx, __shfl_xor(mx, off, 16));
        float sum = 0.f;
        for (int k = t16; k < Kv; k += 16) {
            float e = __expf(prow[k] - mx);
            prow[k] = e;
            sum += e;
        }
        for (int off = 8; off > 0; off >>= 1) sum += __shfl_xor(sum, off, 16);
        if (t16 == 0) s_rowsum[row] = sum;
    }
    __syncthreads();

    // ---- Phase 3: attn = P . V (each wave owns 64 of the 512 Gt cols) -----
    v8f acc[4];
    {
        v8f z = {};
        acc[0] = z; acc[1] = z; acc[2] = z; acc[3] = z;
    }
    const float* Vb = cv + (size_t)b * 512 * 512;
    const int g0 = w << 6;
    for (int kt = 0; kt < nk; ++kt) {
        const int k0 = kt << 4;
#pragma unroll
        for (int j = 0; j < 4; ++j) {
            const int kk = k0 + 4 * j + 2 * h;
            v2f a2 = *(const v2f*)(sm + m * SM_STRIDE + kk);   // P in A-layout
            const float* vr = Vb + (size_t)kk * 512 + g0 + m;  // B: N = m
#pragma unroll
            for (int t = 0; t < 4; ++t) {
                v2f b2;
                b2.x = vr[t * 16];
                b2.y = vr[512 + t * 16];
                acc[t] = wmma4(a2, b2, acc[t]);
            }
        }
    }
    __syncthreads();   // everyone done reading P before we overwrite sm

    // ---- Phase 4a: normalize, add residual, stage y in LDS ----------------
#pragma unroll
    for (int r = 0; r < 8; ++r) {
        const int mr = r + 8 * h;
        const float inv = 1.0f / s_rowsum[mr];
        const float* xrow = x + (size_t)(b * 512 + l0 + mr) * 512;
#pragma unroll
        for (int t = 0; t < 4; ++t) {
            const int g = g0 + t * 16 + m;
            sm[mr * SM_STRIDE + g] = xrow[g] + acc[t][r] * inv;
        }
    }
    __syncthreads();

    // ---- Phase 4b: LayerNorm over Gt = 512 --------------------------------
    {
        const int row = tid >> 4;
        const int t16 = tid & 15;
        const float* yrow = sm + row * SM_STRIDE;
        float s1 = 0.f, s2 = 0.f;
        for (int g = t16; g < 512; g += 16) {
            float v = yrow[g];
            s1 += v; s2 += v * v;
        }
        for (int off = 8; off > 0; off >>= 1) {
            s1 += __shfl_xor(s1, off, 16);
            s2 += __shfl_xor(s2, off, 16);
        }
        const float mean = s1 * (1.0f / 512.0f);
        const float var  = s2 * (1.0f / 512.0f) - mean * mean;
        const float rs   = rsqrtf(var + LN_EPS);
        float* orow = out + (size_t)(b * 512 + l0 + row) * 512;
        for (int g = t16; g < 512; g += 16)
            orow[g] = (yrow[g] - mean) * rs * gamma[g] + beta[g];
    }
}

// ---------------------------------------------------------------------------
extern "C" void kernel_launch(void* const* d_in, const int* in_sizes, int n_in,
                              void* d_out, int out_size, void* d_ws, size_t ws_size,
                              hipStream_t stream) {
    (void)in_sizes; (void)n_in; (void)out_size; (void)ws_size;
    const float* x     = (const float*)d_in[0];  // [4,512,512]
    const float* xq    = (const float*)d_in[1];  // [4,512,512,64]
    const float* ck    = (const float*)d_in[2];  // [4,512,256,64]
    const float* cv    = (const float*)d_in[3];  // [4,512,512]
    const float* gamma = (const float*)d_in[4];  // [512]
    const float* beta  = (const float*)d_in[5];  // [512]
    float* out = (float*)d_out;

    float* qred = (float*)d_ws;                  // [4*512, 64]
    float* kred = qred + (size_t)4 * 512 * 64;   // [4*512, 64]

    reduce_genes<<<4 * 512, 256, 0, stream>>>(xq, qred, 512);
    reduce_genes<<<4 * 512, 256, 0, stream>>>(ck, kred, 256);
    attn_fused<<<4 * 32, 256, 0, stream>>>(x, cv, qred, kred, gamma, beta, out);
}